// ChebNet_WAN_6665789243573
// MI455X (gfx1250) — compile-verified
//
#include <hip/hip_runtime.h>

typedef __attribute__((ext_vector_type(2))) float v2f;
typedef __attribute__((ext_vector_type(8))) float v8f;

#define GPW 16                 // graphs per wave
#define WAVES 4                // waves per block
#define BLOCK_THREADS (WAVES * 32)
#define GPB (WAVES * GPW)      // 64 graphs per block

// per-wave LDS slice layout (floats)
#define XOFF   0
#define XSZ    (GPW * 49)          // 784 : x staging
#define EAOFF  (XOFF + XSZ)
#define EASZ   (GPW * 42)          // 672 : edge_attr staging
#define DISOFF (EAOFF + EASZ)
#define DISSZ  (GPW * 8)           // 128 : dis per node (stride 8, pad slot = 0)
#define H3OFF  (DISOFF + DISSZ)
#define H3SZ   (GPW * 48)          // 768 : flattened h3 (stride 48, pads zeroed)
#define HBOFF  (H3OFF + H3SZ)      // 256 : H tile (16x16)
#define PBOFF  (HBOFF + 256)       // 256 : P tile (16x16)
#define WSLICE (PBOFF + 256)       // 2864 floats per wave

static __device__ __forceinline__ v8f wmma_f32(v2f a, v2f b, v8f c) {
  // D(16x16) = A(16x4) * B(4x16) + C, full fp32 (V_WMMA_F32_16X16X4_F32)
  return __builtin_amdgcn_wmma_f32_16x16x4_f32(
      false, a, false, b, (short)0, c, false, false);
}

__global__ __launch_bounds__(BLOCK_THREADS) void cheb_wmma_kernel(
    const float* __restrict__ x,    const float* __restrict__ ea,
    const float* __restrict__ w10,  const float* __restrict__ w11, const float* __restrict__ b1,
    const float* __restrict__ w20,  const float* __restrict__ w21, const float* __restrict__ b2,
    const float* __restrict__ w30,  const float* __restrict__ w31, const float* __restrict__ b3,
    const float* __restrict__ bng,  const float* __restrict__ bnb,
    const float* __restrict__ bnm,  const float* __restrict__ bnv,
    const float* __restrict__ fcw,  const float* __restrict__ fcb,
    float* __restrict__ out, int num_graphs)
{
  __shared__ float smem[WAVES * WSLICE];
  if ((int)(blockIdx.x * GPB) >= num_graphs) return;

  const int lane = threadIdx.x & 31;
  const int wv   = threadIdx.x >> 5;
  float* ws = &smem[wv * WSLICE];
  const long gb = (long)(blockIdx.x * WAVES + wv) * GPW;  // first graph of this wave

  // ---- stage x and edge_attr (coalesced b32 streams) ----
  {
    const float* xs = x + gb * 49;
    for (int i = lane; i < XSZ; i += 32) ws[XOFF + i] = xs[i];
    const float* es = ea + gb * 42;
    for (int i = lane; i < EASZ; i += 32) ws[EAOFF + i] = es[i];
    // zero h3 pad columns (42..47) so FC A-loads need no guard (and no NaN junk)
    for (int i = lane; i < GPW * 6; i += 32) {
      int g = i / 6, j = i - g * 6;
      ws[H3OFF + g * 48 + 42 + j] = 0.0f;
    }
  }

  // ---- per-lane WMMA operand roles ----
  const int  c  = lane & 15;          // B/D column
  const bool hi = lane >= 16;         // K-half (A/B) ; row+8 (C/D)
  const int  kh = hi ? 2 : 0;
  const int  r  = lane & 15;          // A row
  const int  gr = r >> 3;             // graph-in-pair (row side)
  const int  dn = r & 7;              // dst node (row side)
  const int  rofs = hi ? 8 : 0;       // D row offset

  // weight B-tiles: unconditional clamped-index loads * {0,1} masks (no branches)
  const int   c6 = (c < 6) ? c : 0;
  const float cm = (c < 6) ? 1.0f : 0.0f;
  auto wtile = [&](const float* w, int K, int s) -> v2f {
    int k0 = 4 * s + kh;
    float m0 = (k0     < K) ? cm : 0.0f;
    float m1 = (k0 + 1 < K) ? cm : 0.0f;
    v2f t;
    t.x = w[((k0     < K) ? k0     : 0) * 6 + c6] * m0;
    t.y = w[((k0 + 1 < K) ? k0 + 1 : 0) * 6 + c6] * m1;
    return t;
  };
  v2f Wt0_0[2], Wt1_0[2], Wt0_1[2], Wt1_1[2], Wt0_2[2], Wt1_2[2];
#pragma unroll
  for (int s = 0; s < 2; ++s) {
    Wt0_0[s] = wtile(w10, 7, s);  Wt1_0[s] = wtile(w11, 7, s);
    Wt0_1[s] = wtile(w20, 6, s);  Wt1_1[s] = wtile(w21, 6, s);
    Wt0_2[s] = wtile(w30, 6, s);  Wt1_2[s] = wtile(w31, 6, s);
  }
  v2f FCt[11];
#pragma unroll
  for (int s = 0; s < 11; ++s) {
    int k0 = 4 * s + kh;
    float m0 = (k0     < 42) ? 1.0f : 0.0f;
    float m1 = (k0 + 1 < 42) ? 1.0f : 0.0f;
    FCt[s].x = fcw[((k0     < 42) ? k0     : 0) * 16 + c] * m0;
    FCt[s].y = fcw[((k0 + 1 < 42) ? k0 + 1 : 0) * 16 + c] * m1;
  }
  const float s0  = bng[c6] * rsqrtf(bnv[c6] + 1e-5f);
  const float bnS = s0 * cm;
  const float bnT = (bnb[c6] - bnm[c6] * s0) * cm;
  const float bias0 = b1[c6] * cm;
  const float bias1 = b2[c6] * cm;
  const float bias2 = b3[c6] * cm;

  // ---- lane-constant L-construction tables (independent of pair index) ----
  // Each lane contributes at most 4 nonzero L entries: sn in {kh, kh+1, kh+4, kh+5}
  float lsgn[4];      // -1 if (sn != dn) else 0  (sign of -dis*ea*dis folded in)
  int   eoff[4];      // ea offset within pair slice
  int   doff[4];      // dis offset within pair slice
#pragma unroll
  for (int i = 0; i < 4; ++i) {
    int sn = kh + (i >> 1) * 4 + (i & 1);
    lsgn[i] = (sn != dn) ? -1.0f : 0.0f;
    eoff[i] = gr * 42 + sn * 6 + dn - ((dn > sn) ? 1 : 0);
    doff[i] = gr * 8 + sn;
  }
  const int  ddn  = gr * 8 + dn;
  const bool isg0 = (gr == 0);

  __syncthreads();

  // ---- dis = rsqrt(deg) per node ----
  for (int idx = lane; idx < GPW * 7; idx += 32) {
    int g = idx / 7, n = idx - g * 7;
    float d = 0.0f;
#pragma unroll
    for (int j = 0; j < 6; ++j) d += ws[EAOFF + g * 42 + n * 6 + j];
    ws[DISOFF + g * 8 + n] = (d > 0.0f) ? rsqrtf(d) : 0.0f;
  }
  if (lane < GPW) ws[DISOFF + lane * 8 + 7] = 0.0f;  // pad node -> kills pad terms
  __syncthreads();

  // ---- 8 graph-pairs, each a 16x16 tile ----
  for (int p = 0; p < 8; ++p) {
    const int eb = EAOFF + p * 84;    // pair's edge_attr base
    const int db = DISOFF + p * 16;   // pair's dis base

    // 9 unconditional LDS loads + FMULs; no branches
    const float disd = ws[db + ddn];
    float Lv[4];
#pragma unroll
    for (int i = 0; i < 4; ++i)
      Lv[i] = lsgn[i] * ws[db + doff[i]] * ws[eb + eoff[i]] * disd;

    v2f Lr[4];
    Lr[0].x = isg0 ? Lv[0] : 0.0f;  Lr[0].y = isg0 ? Lv[1] : 0.0f;
    Lr[1].x = isg0 ? Lv[2] : 0.0f;  Lr[1].y = isg0 ? Lv[3] : 0.0f;
    Lr[2].x = isg0 ? 0.0f : Lv[0];  Lr[2].y = isg0 ? 0.0f : Lv[1];
    Lr[3].x = isg0 ? 0.0f : Lv[2];  Lr[3].y = isg0 ? 0.0f : Lv[3];

    v8f hprev;
#pragma unroll
    for (int j = 0; j < 8; ++j) hprev[j] = 0.0f;

#pragma unroll
    for (int layer = 0; layer < 3; ++layer) {
      // A-operand (unconditional; junk lanes hit zero weight rows / dead D slots)
      v2f aH[2];
#pragma unroll
      for (int s = 0; s < 2; ++s) {
        int k0 = 4 * s + kh;
        int base = (layer == 0) ? (XOFF + (p * 2 + gr) * 49 + dn * 7)
                                : (HBOFF + r * 16);
        aH[s].x = ws[base + k0];
        aH[s].y = ws[base + k0 + 1];
      }
      const v2f* W0 = (layer == 0) ? Wt0_0 : (layer == 1) ? Wt0_1 : Wt0_2;
      const v2f* W1 = (layer == 0) ? Wt1_0 : (layer == 1) ? Wt1_1 : Wt1_2;
      const float bias = (layer == 0) ? bias0 : (layer == 1) ? bias1 : bias2;

      v8f d0, pm;
#pragma unroll
      for (int j = 0; j < 8; ++j) { d0[j] = 0.0f; pm[j] = 0.0f; }
#pragma unroll
      for (int s = 0; s < 2; ++s) d0 = wmma_f32(aH[s], W0[s], d0);   // H @ W0
#pragma unroll
      for (int s = 0; s < 2; ++s) pm = wmma_f32(aH[s], W1[s], pm);   // P = H @ W1

      // P (D-layout) -> LDS tile
#pragma unroll
      for (int j = 0; j < 8; ++j) ws[PBOFF + (j + rofs) * 16 + c] = pm[j];
      __syncthreads();

      // d0 += L @ P  (K=16, 4 chained steps)
#pragma unroll
      for (int s = 0; s < 4; ++s) {
        int k0 = 4 * s + kh;
        v2f pb;
        pb.x = ws[PBOFF + k0 * 16 + c];
        pb.y = ws[PBOFF + (k0 + 1) * 16 + c];
        d0 = wmma_f32(Lr[s], pb, d0);
      }

      // relu(z + b) * bnS + bnT (+ 0.7 * residual); cols>=6 stay exact 0
      v8f hnew;
#pragma unroll
      for (int j = 0; j < 8; ++j) {
        float z = d0[j] + bias;
        z = fmaxf(z, 0.0f);
        z = z * bnS + bnT;
        if (layer > 0) z += 0.7f * hprev[j];
        hnew[j] = z;
      }

      if (layer < 2) {
#pragma unroll
        for (int j = 0; j < 8; ++j) ws[HBOFF + (j + rofs) * 16 + c] = hnew[j];
        __syncthreads();
      } else {
        // flatten h3: feat = node*6 + col; guard needed (lanes would collide)
        if (c < 6) {
#pragma unroll
          for (int j = 0; j < 7; ++j)
            ws[H3OFF + (p * 2 + (hi ? 1 : 0)) * 48 + j * 6 + c] = hnew[j];
        }
      }
      hprev = hnew;
    }
  }
  __syncthreads();

  // ---- FC: 16 graphs x 44(pad)  @  44 x 16, 11 chained WMMAs ----
  v8f acc;
#pragma unroll
  for (int j = 0; j < 8; ++j) acc[j] = fcb[c];
#pragma unroll
  for (int s = 0; s < 11; ++s) {
    int k0 = 4 * s + kh;
    v2f a;
    a.x = ws[H3OFF + r * 48 + k0];       // pads zeroed at startup
    a.y = ws[H3OFF + r * 48 + k0 + 1];
    acc = wmma_f32(a, FCt[s], acc);
  }
#pragma unroll
  for (int j = 0; j < 8; ++j) {
    long g = gb + j + rofs;
    out[g * 16 + c] = acc[j];
  }
}

extern "C" void kernel_launch(void* const* d_in, const int* in_sizes, int n_in,
                              void* d_out, int out_size, void* d_ws, size_t ws_size,
                              hipStream_t stream) {
  // 0:x 1:edge_index(unused) 2:edge_attr 3:batch(unused)
  // 4:w1_0 5:w1_1 6:b1 7:w2_0 8:w2_1 9:b2 10:w3_0 11:w3_1 12:b3
  // 13:bn_gamma 14:bn_beta 15:bn_mean 16:bn_var 17:fc_w 18:fc_b
  const float* x   = (const float*)d_in[0];
  const float* ea  = (const float*)d_in[2];
  const float* w10 = (const float*)d_in[4];
  const float* w11 = (const float*)d_in[5];
  const float* b1  = (const float*)d_in[6];
  const float* w20 = (const float*)d_in[7];
  const float* w21 = (const float*)d_in[8];
  const float* b2  = (const float*)d_in[9];
  const float* w30 = (const float*)d_in[10];
  const float* w31 = (const float*)d_in[11];
  const float* b3  = (const float*)d_in[12];
  const float* bng = (const float*)d_in[13];
  const float* bnb = (const float*)d_in[14];
  const float* bnm = (const float*)d_in[15];
  const float* bnv = (const float*)d_in[16];
  const float* fcw = (const float*)d_in[17];
  const float* fcb = (const float*)d_in[18];
  float* out = (float*)d_out;

  const int num_graphs = in_sizes[0] / 49;
  const int blocks = (num_graphs + GPB - 1) / GPB;

  cheb_wmma_kernel<<<blocks, BLOCK_THREADS, 0, stream>>>(
      x, ea, w10, w11, b1, w20, w21, b2, w30, w31, b3,
      bng, bnb, bnm, bnv, fcw, fcb, out, num_graphs);
}